// CenterLoss_51702816309721
// MI455X (gfx1250) — compile-verified
//
#include <hip/hip_runtime.h>

// CenterLoss on MI455X (gfx1250):
//   loss = (1/B) * Σ_b clip(||x_b - centers[labels[b]]||^2, 1e-12, 1e12) + (C-1)*1e-12
// Memory-bound gather (16 MB @ 23.3 TB/s ~ 0.7us). Row sums-of-squares computed on
// the matrix pipe with V_WMMA_F32_16X16X4_F32 via the A*A^T diagonal trick.

#define BATCH 1024
#define NUM_CLASSES 8192
#define FEAT 2048
#define ROWS 16            // WMMA M tile
#define TILE_K 128         // K columns staged per LDS tile
#define KSLICES 4          // K split across blocks for parallelism
#define KPER (FEAT / KSLICES)   // 512
#define PAD 132            // LDS row stride (floats): bank = 4*row + k -> conflict-free

typedef __attribute__((ext_vector_type(2))) float v2f;
typedef __attribute__((ext_vector_type(4))) float v4f;
typedef __attribute__((ext_vector_type(8))) float v8f;

__global__ __launch_bounds__(128) void center_dist_wmma(
    const float* __restrict__ x, const int* __restrict__ labels,
    const float* __restrict__ centers, float* __restrict__ partial) {
  __shared__ __align__(16) float diffT[ROWS * PAD];   // 16 x 128 diff tile, padded
  __shared__ float diag[4 * ROWS];                    // per-wave diagonal partials

  const int t = threadIdx.x;
  const int rowTile = blockIdx.x >> 2;   // 64 row tiles of 16 rows
  const int kslice  = blockIdx.x & 3;    // 4 K slices of 512
  const int kbase0  = kslice * KPER;

  // Staging mapping: 8 threads per row, each loads a float4 from each of 4 32-float segments
  const int rowS = t >> 3;
  const int l8   = t & 7;
  const int gRow = rowTile * ROWS + rowS;
  const float* xr = x + (long long)gRow * FEAT;
  const float* cr = centers + (long long)labels[gRow] * FEAT;

  // WMMA f32 16x16x4 fragment mapping: lane m<16 -> row m, K{0,1};
  // lane m+16 -> row m, K{2,3}. B(4x16) layout coincides, so passing the same
  // registers as A and B computes D = A * A^T (diag = row sum of squares).
  const int l    = t & 31;
  const int w    = t >> 5;               // wave 0..3 owns 32 K-cols of the tile
  const int rowA = l & 15;
  const int koff = (l >> 4) << 1;        // 0 or 2

  v8f acc = {};

  for (int tile = 0; tile < KPER / TILE_K; ++tile) {
    const int kb = kbase0 + tile * TILE_K;
    // Stage diff = x - gathered center (coalesced 128B row segments -> LDS)
#pragma unroll
    for (int p = 0; p < 4; ++p) {
      const int col = p * 32 + l8 * 4;
      v4f xv = *(const v4f*)(xr + kb + col);
      v4f cv = *(const v4f*)(cr + kb + col);
      v4f d  = xv - cv;
      *(v4f*)(&diffT[rowS * PAD + col]) = d;
    }
    if (tile < KPER / TILE_K - 1) {   // gfx1250 global_prefetch_b8 for next tile
      __builtin_prefetch(xr + kb + TILE_K + l8 * 16, 0, 0);
      __builtin_prefetch(cr + kb + TILE_K + l8 * 16, 0, 0);
    }
    __syncthreads();

    // Each wave: 8 chained f32 WMMAs over its 32-column K slice
    const float* abase = &diffT[rowA * PAD + w * 32 + koff];
#pragma unroll
    for (int ks = 0; ks < 8; ++ks) {
      v2f a = *(const v2f*)(abase + ks * 4);   // ds_load_b64, conflict-free
      acc = __builtin_amdgcn_wmma_f32_16x16x4_f32(
          false, a, false, a, (short)0, acc, false, false);
    }
    __syncthreads();
  }

  // Extract diagonal: rows 0-7 live at lane r, acc[r]; rows 8-15 at lane r+8, acc[r-8]
  float val = 0.f;
  int myrow = -1;
#pragma unroll
  for (int j = 0; j < 8; ++j) {
    if (l == j)      { val = acc[j]; myrow = j; }
    if (l == 16 + j) { val = acc[j]; myrow = 8 + j; }
  }
  if (myrow >= 0) diag[w * ROWS + myrow] = val;
  __syncthreads();

  if (t < ROWS) {
    float dsum = diag[t] + diag[ROWS + t] + diag[2 * ROWS + t] + diag[3 * ROWS + t];
    partial[kslice * BATCH + rowTile * ROWS + t] = dsum;
  }
}

__global__ __launch_bounds__(256) void center_loss_finalize(
    const float* __restrict__ partial, float* __restrict__ out) {
  __shared__ float red[256];
  const int t = threadIdx.x;
  float s = 0.f;
#pragma unroll
  for (int i = 0; i < 4; ++i) {
    const int row = t * 4 + i;   // 0..1023
    float d = partial[row] + partial[BATCH + row] + partial[2 * BATCH + row] +
              partial[3 * BATCH + row];
    d = fminf(fmaxf(d, 1e-12f), 1e12f);   // clip applied to true-class entries
    s += d;
  }
  red[t] = s;
  __syncthreads();
  for (int off = 128; off > 0; off >>= 1) {
    if (t < off) red[t] += red[t + off];
    __syncthreads();
  }
  if (t == 0) {
    // masked-out zeros each contribute clip(0)=1e-12: (B*C - B)*1e-12 / B = (C-1)*1e-12
    out[0] = red[0] * (1.0f / (float)BATCH) + (float)(NUM_CLASSES - 1) * 1e-12f;
  }
}

extern "C" void kernel_launch(void* const* d_in, const int* in_sizes, int n_in,
                              void* d_out, int out_size, void* d_ws, size_t ws_size,
                              hipStream_t stream) {
  const float* x       = (const float*)d_in[0];   // (1024, 2048) f32
  const int*   labels  = (const int*)d_in[1];     // (1024,) int
  const float* centers = (const float*)d_in[2];   // (8192, 2048) f32
  float* out     = (float*)d_out;                 // scalar loss
  float* partial = (float*)d_ws;                  // 4 * 1024 floats of scratch

  center_dist_wmma<<<dim3(64 * KSLICES), dim3(128), 0, stream>>>(x, labels, centers,
                                                                 partial);
  center_loss_finalize<<<dim3(1), dim3(256), 0, stream>>>(partial, out);
}